// EdgePredictor_7215545058050
// MI455X (gfx1250) — compile-verified
//
#include <hip/hip_runtime.h>
#include <cstddef>

typedef __attribute__((ext_vector_type(2))) float v2f;
typedef __attribute__((ext_vector_type(4))) float v4f;
typedef __attribute__((ext_vector_type(8))) float v8f;

// ---------------------------------------------------------------------------
// Pass 1: zero the dense [N,N] output with non-temporal 128-bit stores.
// 604 MB >> 192 MB L2, so NT (bypass/rinse) is the right temporal hint.
// ---------------------------------------------------------------------------
__global__ void ep_zero_kernel(v4f* __restrict__ out, size_t n4) {
    size_t i = (size_t)blockIdx.x * blockDim.x + threadIdx.x;
    if (i < n4) {
        v4f zv = {0.f, 0.f, 0.f, 0.f};
        __builtin_nontemporal_store(zv, out + i);
    }
}

// ---------------------------------------------------------------------------
// Pass 2: per-node dots  a[n] = z[n]·W[0:128],  c[n] = z[n]·W[128:256]
// as a GEMM  z[N x 128] @ V[128 x 16]  (V cols 0,1 = W halves, rest zero)
// using V_WMMA_F32_16X16X4_F32.  One wave per 16-row tile, K in steps of 4.
//
// A 16x4 f32 layout: lane<16 -> M=lane,  K={0,1} in v0,v1
//                    lane>=16 -> M=lane-16, K={2,3}
// B 4x16  f32 layout: lane<16 -> N=lane,  K={0,1}; lane>=16 -> N=lane-16, K={2,3}
// D 16x16 f32 layout: VGPR v, lane<16 -> M=v, lane>=16 -> M=v+8, N=lane&15
//
// B loads are branch-free: every lane loads from a safe in-bounds address
// ((mn&1)*latent stays inside W's 256 floats) and lanes with N>=2 multiply
// by 0 — avoids per-iteration exec save/restore around the loads so the
// compiler can pipeline loads against WMMA issue.
// ---------------------------------------------------------------------------
__global__ __launch_bounds__(32) void ep_node_dot_wmma(
        const float* __restrict__ z, const float* __restrict__ W,
        float* __restrict__ ac, int latent) {
    const int lane  = threadIdx.x;
    const int mn    = lane & 15;          // row of A / col of B held by this lane
    const int khalf = (lane >> 4) << 1;   // 0 or 2: which K-pair this lane holds

    const float* zrow = z + (size_t)(blockIdx.x * 16 + mn) * (size_t)latent;
    const float* wrow = W + (mn & 1) * latent + khalf;   // safe for all lanes
    const float bmask = (mn < 2) ? 1.0f : 0.0f;          // only B cols 0,1 nonzero

    v8f acc = {};
    for (int k0 = 0; k0 < latent; k0 += 4) {
        v2f a;
        a.x = zrow[k0 + khalf];
        a.y = zrow[k0 + khalf + 1];
        v2f bm;
        bm.x = wrow[k0]     * bmask;
        bm.y = wrow[k0 + 1] * bmask;
        acc = __builtin_amdgcn_wmma_f32_16x16x4_f32(
            /*neg_a=*/false, a, /*neg_b=*/false, bm,
            /*c_mod=*/(short)0, acc, /*reuse_a=*/false, /*reuse_b=*/false);
    }

    if (mn < 2) {                         // lanes holding columns 0 (=a) and 1 (=c)
        int mbase = blockIdx.x * 16 + ((lane >> 4) << 3);
#pragma unroll
        for (int v = 0; v < 8; ++v)
            ac[(size_t)(mbase + v) * 2 + mn] = acc[v];
    }
}

// ---------------------------------------------------------------------------
// Pass 3: scatter-SET edge probabilities:  out[row,col] = a[row]+c[col]+b
// atomicExch == indexed assignment (one winner among duplicate keys).
// ---------------------------------------------------------------------------
__global__ void ep_edge_prob_set(const int* __restrict__ ei,
                                 const float* __restrict__ ac,
                                 const float* __restrict__ bias,
                                 float* __restrict__ out,
                                 int n_edges, int n_nodes) {
    int e = blockIdx.x * blockDim.x + threadIdx.x;
    if (e >= n_edges) return;
    int r = ei[e];
    int c = ei[n_edges + e];
    float p = ac[(size_t)r * 2 + 0] + ac[(size_t)c * 2 + 1] + bias[0];
    atomicExch(&out[(size_t)r * n_nodes + c], p);
}

// ---------------------------------------------------------------------------
// Pass 4: scatter-ADD summed edge attributes: out[row,col] += sum(edge_attr[e])
// Runs strictly after pass 3 (stream order), matching set-then-add semantics.
// ---------------------------------------------------------------------------
__global__ void ep_edge_attr_add(const int* __restrict__ ei,
                                 const float* __restrict__ ea,
                                 float* __restrict__ out,
                                 int n_edges, int n_nodes) {
    int e = blockIdx.x * blockDim.x + threadIdx.x;
    if (e >= n_edges) return;
    int r = ei[e];
    int c = ei[n_edges + e];
    v4f a = *(const v4f*)(ea + (size_t)e * 4);   // EDGE_F == 4, 16B aligned
    float s = (a.x + a.y) + (a.z + a.w);
    atomicAdd(&out[(size_t)r * n_nodes + c], s);
}

extern "C" void kernel_launch(void* const* d_in, const int* in_sizes, int n_in,
                              void* d_out, int out_size, void* d_ws, size_t ws_size,
                              hipStream_t stream) {
    const float* z  = (const float*)d_in[0];   // [N, 128]
    const int*   ei = (const int*)d_in[1];     // [2, E]
    const float* ea = (const float*)d_in[2];   // [E, 4]
    const float* W  = (const float*)d_in[3];   // [1, 256]
    const float* b  = (const float*)d_in[4];   // [1]
    float* out = (float*)d_out;                // [N, N]

    const int latent  = in_sizes[3] / 2;       // 128
    const int n_nodes = in_sizes[0] / latent;  // 12288
    const int n_edges = in_sizes[1] / 2;       // 393216

    float* ac = (float*)d_ws;                  // [N, 2] : (a[n], c[n]) -- 96 KB

    // 1) zero dense output (bandwidth-dominant pass: ~604 MB NT stores)
    size_t n4 = (size_t)out_size / 4;
    int zb = (int)((n4 + 255) / 256);
    ep_zero_kernel<<<zb, 256, 0, stream>>>((v4f*)out, n4);

    // 2) per-node dot products via f32 WMMA (768 tiles of 16 rows)
    ep_node_dot_wmma<<<n_nodes / 16, 32, 0, stream>>>(z, W, ac, latent);

    // 3) scatter-set probs, then 4) scatter-add attr sums (stream-ordered)
    int eb = (n_edges + 255) / 256;
    ep_edge_prob_set<<<eb, 256, 0, stream>>>(ei, ac, b, out, n_edges, n_nodes);
    ep_edge_attr_add<<<eb, 256, 0, stream>>>(ei, ea, out, n_edges, n_nodes);
}